// KvPageCache_43319040147648
// MI455X (gfx1250) — compile-verified
//
#include <hip/hip_runtime.h>
#include <stdint.h>

// ---------------------------------------------------------------------------
// Paged KV-cache scatter write for MI455X (gfx1250).
//   out = copy(kv_pages); out[page,slot,0::2,:] = new_k; out[page,slot,1::2,:] = new_v
// Memory-bound. Bulk copy streams through LDS with the gfx1250 async engine
// (GLOBAL_LOAD_ASYNC_TO_LDS_B128 / GLOBAL_STORE_ASYNC_FROM_LDS_B128,
// ASYNCcnt), ping-pong double buffered. A dirty-slab mask skips copying the
// 8KB slabs that the scatter fully overwrites (~11% traffic saved).
// Launch order on stream: zero_mask -> mark_dirty -> copy -> scatter.
// ---------------------------------------------------------------------------

#define BLK 256   // 8 waves (wave32)
#define UB  8     // async batch depth per buffer; tile = BLK*UB float4 = 32KB = 4 slabs

__device__ __forceinline__ void async_load_b128(unsigned lds_off, unsigned long long gaddr) {
    asm volatile("global_load_async_to_lds_b128 %0, %1, off"
                 :: "v"(lds_off), "v"(gaddr) : "memory");
}
__device__ __forceinline__ void async_store_b128(unsigned long long gaddr, unsigned lds_off) {
    asm volatile("global_store_async_from_lds_b128 %0, %1, off"
                 :: "v"(gaddr), "v"(lds_off) : "memory");
}
__device__ __forceinline__ void wait_async0() {
    asm volatile("s_wait_asynccnt 0" ::: "memory");
}

// --------------------------------------------------------------------------
// Zero the dirty mask (nwords u32 words). Small; avoids hipMemsetAsync.
// --------------------------------------------------------------------------
__global__ void __launch_bounds__(BLK)
zero_mask(unsigned* __restrict__ w, int nwords) {
    for (int i = blockIdx.x * BLK + threadIdx.x; i < nwords; i += gridDim.x * BLK)
        w[i] = 0u;
}

// --------------------------------------------------------------------------
// Mark slabs that the scatter will fully overwrite. Guard matches scatter's
// drop semantics exactly (marked => scatter rewrites the whole 8KB slab).
// --------------------------------------------------------------------------
__global__ void __launch_bounds__(BLK)
mark_dirty(const int* __restrict__ tp, const int* __restrict__ ts,
           unsigned char* __restrict__ mask, int ntok, int num_pages) {
    int i = blockIdx.x * BLK + threadIdx.x;
    if (i < ntok) {
        int p = tp[i], s = ts[i];
        if ((unsigned)p < (unsigned)num_pages && (unsigned)s < 16u)
            mask[p * 16 + s] = 1;   // benign write races
    }
}

// ---------------------------------------------------------------------------
// Kernel: copy kv_pages -> d_out through LDS with the async engine.
// Tile = 2048 float4 (32KB) = slabs [4*tile .. 4*tile+3]; one uniform u32
// mask load per tile, byte b = slab 4*tile+b. Async chunk u (4KB) lies
// entirely in slab 4*tile+(u>>1); skip it when that slab is dirty.
// Ping-pong buffers: tile i+1's loads overlap tile i's stores, so each
// s_wait_asynccnt 0 has up to 16 B128 ops (8KB/wave) in flight.
// ---------------------------------------------------------------------------
__global__ void __launch_bounds__(BLK)
copy_pages_async(const float4* __restrict__ src, float4* __restrict__ dst,
                 long long n4, const unsigned* __restrict__ mask) {
    __shared__ __align__(16) char stage[2][UB][BLK][16];   // 64KB LDS

    const int tid = threadIdx.x;
    const long long tile_elems = (long long)UB * BLK;      // 2048 float4
    const long long ntiles     = n4 / tile_elems;

    int buf = 0;
    for (long long tile = blockIdx.x; tile < ntiles; tile += gridDim.x) {
        const unsigned  m    = mask ? mask[tile] : 0u;     // 4 dirty-slab bytes (uniform)
        const long long base = tile * tile_elems + tid;
        unsigned loff[UB];
        if (m == 0u) {
            // fast path: whole tile clean
#pragma unroll
            for (int u = 0; u < UB; ++u) {
                loff[u] = (unsigned)(uintptr_t)&stage[buf][u][tid][0];
                async_load_b128(loff[u], (unsigned long long)(uintptr_t)(src + base + (long long)u * BLK));
            }
            wait_async0();
#pragma unroll
            for (int u = 0; u < UB; ++u)
                async_store_b128((unsigned long long)(uintptr_t)(dst + base + (long long)u * BLK), loff[u]);
        } else {
            // partial tile: issue only chunks whose slab is clean (uniform branches)
#pragma unroll
            for (int u = 0; u < UB; ++u) {
                if (((m >> ((u >> 1) * 8)) & 0xFFu) == 0u) {
                    loff[u] = (unsigned)(uintptr_t)&stage[buf][u][tid][0];
                    async_load_b128(loff[u], (unsigned long long)(uintptr_t)(src + base + (long long)u * BLK));
                }
            }
            wait_async0();
#pragma unroll
            for (int u = 0; u < UB; ++u) {
                if (((m >> ((u >> 1) * 8)) & 0xFFu) == 0u) {
                    unsigned lo = (unsigned)(uintptr_t)&stage[buf][u][tid][0];
                    async_store_b128((unsigned long long)(uintptr_t)(dst + base + (long long)u * BLK), lo);
                }
            }
        }
        buf ^= 1;                            // next tile's loads use the other buffer
    }

    // tail (n4 not a multiple of the tile size; not hit for this shape)
    for (long long i = ntiles * tile_elems + (long long)blockIdx.x * BLK + tid;
         i < n4; i += (long long)gridDim.x * BLK) {
        dst[i] = src[i];
    }
    wait_async0();                           // ensure all async stores retired
}

// ---------------------------------------------------------------------------
// Scatter new_k/new_v into d_out (runs last; correctness independent of mask).
// One float4 per thread: g -> token t = g>>9, kv_row j = (g>>5)&15, col4 = g&31.
// Reads fully coalesced; each (t,j) writes a contiguous 512B run.
// ---------------------------------------------------------------------------
__global__ void __launch_bounds__(BLK)
scatter_kv(const float4* __restrict__ nk, const float4* __restrict__ nv,
           const int* __restrict__ tp, const int* __restrict__ ts,
           float4* __restrict__ out, int ntok, int num_pages) {
    const long long total = (long long)ntok * 512;   // 16 rows * 32 float4 per token
    for (long long g = (long long)blockIdx.x * BLK + threadIdx.x;
         g < total; g += (long long)gridDim.x * BLK) {
        const int t = (int)(g >> 9);
        const int r = (int)(g & 511);
        const int j = r >> 5;                 // kv row 0..15 (even<-K, odd<-V)
        const int c = r & 31;                 // float4 column 0..31
        const int page = tp[t];
        const int slot = ts[t];
        if ((unsigned)page < (unsigned)num_pages && (unsigned)slot < 16u) {
            const float4* __restrict__ s = (j & 1) ? nv : nk;
            const long long sidx = (long long)t * 256 + (long long)(j >> 1) * 32 + c;
            const long long didx = ((((long long)page * 16 + slot) * 16) + j) * 32 + c;
            out[didx] = s[sidx];
        }
    }
}

extern "C" void kernel_launch(void* const* d_in, const int* in_sizes, int n_in,
                              void* d_out, int out_size, void* d_ws, size_t ws_size,
                              hipStream_t stream) {
    const float* kv = (const float*)d_in[0];
    const float* nk = (const float*)d_in[1];
    const float* nv = (const float*)d_in[2];
    const int*   tp = (const int*)d_in[3];
    const int*   ts = (const int*)d_in[4];
    float*      out = (float*)d_out;

    const long long n_elems   = (long long)in_sizes[0];        // 134,217,728 floats
    const long long n4        = n_elems / 4;                    // float4 count
    const int       ntok      = in_sizes[3];                    // 8192
    const int       num_pages = (int)(n_elems / (16LL * 16 * 128));

    // Dirty-slab mask lives in d_ws: one byte per (page,slot) slab.
    const size_t mask_bytes = (size_t)num_pages * 16;
    const bool   use_mask   = (ws_size >= mask_bytes) && (n4 % ((long long)UB * BLK) == 0);
    unsigned*    mask_w     = use_mask ? (unsigned*)d_ws : nullptr;

    if (use_mask) {
        const int nwords = (int)(mask_bytes / 4);
        zero_mask <<<64, BLK, 0, stream>>>(mask_w, nwords);
        mark_dirty<<<(ntok + BLK - 1) / BLK, BLK, 0, stream>>>(tp, ts,
                                                               (unsigned char*)d_ws,
                                                               ntok, num_pages);
    }

    // Bulk copy (skips slabs the scatter will rewrite).
    copy_pages_async<<<4096, BLK, 0, stream>>>((const float4*)kv, (float4*)out,
                                               n4, (const unsigned*)mask_w);

    // Scatter K/V rows last (overwrites any slab it marked).
    const long long total4 = (long long)ntok * 512;
    int sgrid = (int)((total4 + BLK - 1) / BLK);
    if (sgrid > 16384) sgrid = 16384;
    scatter_kv<<<sgrid, BLK, 0, stream>>>((const float4*)nk, (const float4*)nv,
                                          tp, ts, (float4*)out, ntok, num_pages);
}